// InterpretableAttention_20126216750030
// MI455X (gfx1250) — compile-verified
//
#include <hip/hip_runtime.h>
#include <math.h>

// Problem constants (reference: B=8, N=4096, DIM=1024)
#define NB   8
#define NN   4096
#define ND   1024
#define ECHUNKS 16                 // e-dimension tiles for vproj
#define ECLEN  (ND / ECHUNKS)      // 64 e-values per chunk

typedef __attribute__((ext_vector_type(2))) float v2f;
typedef __attribute__((ext_vector_type(8))) float v8f;

// -----------------------------------------------------------------------------
// Stage 1: q0[b][e] = sum_d Wq[e,d] * x[b,0,d] + bq[e]
// One wave per output row e (coalesced Wq row reads), x[b,0,:] staged in LDS.
// grid = 128 blocks * 8 waves = 1024 rows. Reads Wq exactly once (4 MB).
// -----------------------------------------------------------------------------
__global__ void qrow_kernel(const float* __restrict__ x,
                            const float* __restrict__ Wq,
                            const float* __restrict__ bq,
                            float* __restrict__ q0) {
    __shared__ float sx[NB][ND];            // 32 KB
    const int tid = threadIdx.x;            // 0..255
    for (int i = tid; i < NB * ND; i += 256) {
        int b = i >> 10, d = i & (ND - 1);
        sx[b][d] = x[(size_t)b * NN * ND + d];   // row n=0 of batch b
    }
    __syncthreads();

    const int wave = tid >> 5;
    const int lane = tid & 31;
    const int e = blockIdx.x * 8 + wave;
    const float* __restrict__ wrow = Wq + (size_t)e * ND;

    float acc[NB];
#pragma unroll
    for (int b = 0; b < NB; ++b) acc[b] = 0.f;

    for (int d = lane; d < ND; d += 32) {
        float w = wrow[d];                  // coalesced 128B per wave
#pragma unroll
        for (int b = 0; b < NB; ++b) acc[b] += w * sx[b][d];
    }

#pragma unroll
    for (int b = 0; b < NB; ++b) {
        float r = acc[b];
#pragma unroll
        for (int off = 16; off > 0; off >>= 1) r += __shfl_xor(r, off, 32);
        if (lane == 0) q0[b * ND + e] = r + bq[e];
    }
}

// -----------------------------------------------------------------------------
// Stage 2a: partial v: part[ec][b][d] = sum_{e in chunk ec} Wk[e,d]*q0[b][e]
// grid = (ND/256 d-blocks, ECHUNKS e-chunks) = 64 blocks; each block reads a
// disjoint 64KB slab of Wk -> Wk streamed exactly once at full parallelism.
// -----------------------------------------------------------------------------
__global__ void vproj_partial_kernel(const float* __restrict__ Wk,
                                     const float* __restrict__ q0,
                                     float* __restrict__ part) {
    __shared__ float sq[NB][ECLEN];         // 2 KB
    const int tid = threadIdx.x;
    const int ec  = blockIdx.y;
    const int e0  = ec * ECLEN;
    for (int i = tid; i < NB * ECLEN; i += 256) {
        int b = i / ECLEN, e = i % ECLEN;
        sq[b][e] = q0[b * ND + e0 + e];
    }
    __syncthreads();

    const int d = blockIdx.x * 256 + tid;
    float acc[NB];
#pragma unroll
    for (int b = 0; b < NB; ++b) acc[b] = 0.f;

    const float* __restrict__ col = Wk + (size_t)e0 * ND + d;
    for (int e = 0; e < ECLEN; ++e) {
        float w = col[(size_t)e * ND];      // coalesced across threads
#pragma unroll
        for (int b = 0; b < NB; ++b) acc[b] += w * sq[b][e];   // LDS broadcast
    }
#pragma unroll
    for (int b = 0; b < NB; ++b)
        part[((size_t)ec * NB + b) * ND + d] = acc[b];
}

// -----------------------------------------------------------------------------
// Stage 2b: deterministic reduction of the 16 partials -> v[b][d].
// grid = 32 blocks * 256 threads = 8192 outputs.
// -----------------------------------------------------------------------------
__global__ void vreduce_kernel(const float* __restrict__ part,
                               float* __restrict__ v) {
    const int i = blockIdx.x * 256 + threadIdx.x;   // i = b*ND + d
    float s = 0.f;
#pragma unroll
    for (int ec = 0; ec < ECHUNKS; ++ec) s += part[(size_t)ec * NB * ND + i];
    v[i] = s;
}

// -----------------------------------------------------------------------------
// Stage 3 (dominant, HBM-bound: streams x once = 134 MB ~ 5.8 us roofline):
//   scores[b][m] = dot(v[b], x[b,m,:])  via V_WMMA_F32_16X16X4_F32.
// Each wave: 16 rows of x as the A 16x4 tile; B = v[k..k+3] replicated across
// all 16 columns. 4 accumulators keep the XDL RAW distance at 4 while 16 b64
// loads per compiled iteration stay in flight.
// grid = (NN/128, NB), block = 256 (8 waves * 16 rows).
// -----------------------------------------------------------------------------
__global__ void scores_kernel(const float* __restrict__ x,
                              const float* __restrict__ v,
                              float* __restrict__ scores) {
    const int lane = threadIdx.x & 31;
    const int wave = threadIdx.x >> 5;
    const int b = blockIdx.y;
    const int m0 = (blockIdx.x * 8 + wave) * 16;

    // A layout (16x4 f32): lanes 0-15 -> (M=lane, K=0,1); lanes 16-31 -> (M=lane-16, K=2,3)
    const int row   = m0 + (lane & 15);
    const int kLane = (lane >> 4) * 2;      // 0 or 2

    const float* __restrict__ ap = x + ((size_t)b * NN + row) * ND + kLane;
    const float* __restrict__ bp = v + b * ND + kLane;   // L0-resident, 4 KB

    v8f acc0 = {}, acc1 = {}, acc2 = {}, acc3 = {};
#pragma unroll 2
    for (int k = 0; k < ND; k += 16) {
        v2f a0 = *(const v2f*)(ap + k);
        v2f b0 = *(const v2f*)(bp + k);
        v2f a1 = *(const v2f*)(ap + k + 4);
        v2f b1 = *(const v2f*)(bp + k + 4);
        v2f a2 = *(const v2f*)(ap + k + 8);
        v2f b2 = *(const v2f*)(bp + k + 8);
        v2f a3 = *(const v2f*)(ap + k + 12);
        v2f b3 = *(const v2f*)(bp + k + 12);
        acc0 = __builtin_amdgcn_wmma_f32_16x16x4_f32(false, a0, false, b0,
                                                     (short)0, acc0, false, false);
        acc1 = __builtin_amdgcn_wmma_f32_16x16x4_f32(false, a1, false, b1,
                                                     (short)0, acc1, false, false);
        acc2 = __builtin_amdgcn_wmma_f32_16x16x4_f32(false, a2, false, b2,
                                                     (short)0, acc2, false, false);
        acc3 = __builtin_amdgcn_wmma_f32_16x16x4_f32(false, a3, false, b3,
                                                     (short)0, acc3, false, false);
    }
    v8f acc = (acc0 + acc1) + (acc2 + acc3);

    // D layout: VGPR j -> lanes 0-15: M=j ; lanes 16-31: M=j+8 (all N cols equal)
    if (lane == 0) {
#pragma unroll
        for (int j = 0; j < 8; ++j) scores[(size_t)b * NN + m0 + j] = acc[j];
    } else if (lane == 16) {
#pragma unroll
        for (int j = 0; j < 8; ++j) scores[(size_t)b * NN + m0 + 8 + j] = acc[j];
    }
}

// -----------------------------------------------------------------------------
// Stage 4: row softmax over 4096 scores per batch. One block per batch.
// (The dropped q0.bk constant shifts every score equally -> softmax invariant.)
// -----------------------------------------------------------------------------
__global__ void softmax_kernel(const float* __restrict__ scores,
                               float* __restrict__ out) {
    __shared__ float red[256];
    const int b = blockIdx.x, tid = threadIdx.x;
    const float* __restrict__ s = scores + (size_t)b * NN;

    float m = -INFINITY;
    for (int i = tid; i < NN; i += 256) m = fmaxf(m, s[i]);
    red[tid] = m; __syncthreads();
    for (int off = 128; off > 0; off >>= 1) {
        if (tid < off) red[tid] = fmaxf(red[tid], red[tid + off]);
        __syncthreads();
    }
    const float mx = red[0];
    __syncthreads();

    float sum = 0.f;
    for (int i = tid; i < NN; i += 256) sum += __expf(s[i] - mx);
    red[tid] = sum; __syncthreads();
    for (int off = 128; off > 0; off >>= 1) {
        if (tid < off) red[tid] += red[tid + off];
        __syncthreads();
    }
    const float inv = 1.f / red[0];

    for (int i = tid; i < NN; i += 256)
        out[(size_t)b * NN + i] = __expf(s[i] - mx) * inv;
}

// -----------------------------------------------------------------------------
extern "C" void kernel_launch(void* const* d_in, const int* in_sizes, int n_in,
                              void* d_out, int out_size, void* d_ws, size_t ws_size,
                              hipStream_t stream) {
    const float* x  = (const float*)d_in[0];   // [8,4096,1024]
    const float* Wq = (const float*)d_in[1];   // [1024,1024]
    const float* bq = (const float*)d_in[2];   // [1024]
    const float* Wk = (const float*)d_in[3];   // [1024,1024]
    const float* bk = (const float*)d_in[4];   // unused: softmax shift-invariance
    (void)bk; (void)in_sizes; (void)n_in; (void)out_size; (void)ws_size;

    float* ws   = (float*)d_ws;
    float* q0   = ws;                              // 8*1024 floats
    float* vv   = ws + NB * ND;                    // 8*1024 floats
    float* sc   = ws + 2 * NB * ND;                // 8*4096 floats
    float* part = ws + 2 * NB * ND + NB * NN;      // 16*8*1024 floats (512 KB)

    qrow_kernel         <<<dim3(ND / 8),            256, 0, stream>>>(x, Wq, bq, q0);
    vproj_partial_kernel<<<dim3(ND / 256, ECHUNKS), 256, 0, stream>>>(Wk, q0, part);
    vreduce_kernel      <<<dim3(NB * ND / 256),     256, 0, stream>>>(part, vv);
    scores_kernel       <<<dim3(NN / 128, NB),      256, 0, stream>>>(x, vv, sc);
    softmax_kernel      <<<dim3(NB),                256, 0, stream>>>(sc, (float*)d_out);
}